// FastGuidedFilter_57930518889198
// MI455X (gfx1250) — compile-verified
//
#include <hip/hip_runtime.h>
#include <stdint.h>

typedef float v2f __attribute__((ext_vector_type(2)));
typedef float v8f __attribute__((ext_vector_type(8)));

namespace {
constexpr int R    = 8;      // box radius (window = 17)
constexpr int TILE = 64;     // output tile per block
constexpr int XT   = 96;     // x/y tile extent = TILE + 4R
constexpr int XS   = 100;    // LDS stride for x/y (float4-aligned, conflict-free)
constexpr int HT   = 96;     // H-buffer rows
constexpr int HC   = 80;     // H-buffer cols = TILE + 2R
constexpr int HS   = 80;
constexpr int AT   = 80;     // A/b extent
constexpr int AS   = 82;     // A/b stride (even, conflict-free column reads)
constexpr int H2C  = 64;
constexpr int H2S  = 64;
constexpr float EPS = 1e-8f;

// LDS layout (floats)
constexpr int OFF_SX = 0;
constexpr int OFF_SY = OFF_SX + XT * XS;
constexpr int OFF_H0 = OFF_SY + XT * XS;          // 4 buffers of HT*HS
constexpr int OFF_SA = OFF_H0 + 4 * HT * HS;
constexpr int OFF_SB = OFF_SA + AT * AS;
constexpr int OFF_HA = OFF_SB + AT * AS;
constexpr int OFF_HB = OFF_HA + AT * H2S;
constexpr int SMEM_FLOATS = OFF_HB + AT * H2S;    // 73280 floats = 293120 B
} // namespace

// band weight: 1 if 0 <= k-j <= 2R  (window of 17 taps)
__device__ __forceinline__ float bandw(int k, int j) {
  unsigned d = (unsigned)(k - j);
  return (d <= 2u * R) ? 1.0f : 0.0f;
}

// truncated-window pixel count along one axis (matches reference boxfilter(ones))
__device__ __forceinline__ float cntf(int i, int n) {
  int lo = i - R; lo = lo < 0 ? 0 : lo;
  int hi = i + R; hi = hi > n - 1 ? n - 1 : hi;
  return (float)(hi - lo + 1);
}

__device__ __forceinline__ v8f wmma_f32(v2f a, v2f b, v8f c) {
  // D = A(16x4) * B(4x16) + C ; emits v_wmma_f32_16x16x4_f32
  return __builtin_amdgcn_wmma_f32_16x16x4_f32(false, a, false, b, (short)0, c,
                                               false, false);
}

// Async global->LDS 16-byte copy (GLOBAL_LOAD_ASYNC_TO_LDS_B128, ASYNCcnt).
// lds_addr: low 32 bits of the shared-aperture pointer == LDS byte address.
__device__ __forceinline__ void async_copy_b128(unsigned lds_addr,
                                                const void* gptr) {
  uint64_t ga = (uint64_t)(uintptr_t)gptr;
  asm volatile("global_load_async_to_lds_b128 %0, %1, off"
               :
               : "v"(lds_addr), "v"(ga)
               : "memory");
}

__device__ __forceinline__ void wait_async_zero() {
  asm volatile("s_wait_asynccnt 0x0" ::: "memory");
}

// Horizontal box sums: 16 rows (r0..r0+15) of `src` (stride ss); output col j
// (j=0..15, at v0+j) = sum of src cols v0+j .. v0+j+16.  Data is the A matrix,
// 0/1 band is the B matrix.  8 chained K=4 WMMAs.
__device__ __forceinline__ v8f box_h(const float* src, int ss, int r0, int v0,
                                     int lane) {
  const int M   = lane & 15;
  const int ko  = (lane < 16) ? 0 : 2;   // A: VGPR0 holds K={0,2}, VGPR1 K={1,3}
  const int kb0 = (lane < 16) ? 0 : 1;   // B: VGPR0 holds K={0,1}, VGPR1 K={2,3}
  const float* row = src + (r0 + M) * ss;
  v8f acc = {0.f, 0.f, 0.f, 0.f, 0.f, 0.f, 0.f, 0.f};
#pragma unroll
  for (int kb = 0; kb < 8; ++kb) {
    const int u = v0 + kb * 4 + ko;
    v2f a = {row[u], row[u + 1]};
    v2f b = {bandw(kb * 4 + kb0, M), bandw(kb * 4 + kb0 + 2, M)};
    acc = wmma_f32(a, b, acc);
  }
  return acc;
}

// Same, but A-matrix entries are products s1*s2 (for boxfilter(x*y), (x*x))
__device__ __forceinline__ v8f box_h_prod(const float* s1, const float* s2,
                                          int ss, int r0, int v0, int lane) {
  const int M   = lane & 15;
  const int ko  = (lane < 16) ? 0 : 2;
  const int kb0 = (lane < 16) ? 0 : 1;
  const float* r1 = s1 + (r0 + M) * ss;
  const float* r2 = s2 + (r0 + M) * ss;
  v8f acc = {0.f, 0.f, 0.f, 0.f, 0.f, 0.f, 0.f, 0.f};
#pragma unroll
  for (int kb = 0; kb < 8; ++kb) {
    const int u = v0 + kb * 4 + ko;
    v2f a = {r1[u] * r2[u], r1[u + 1] * r2[u + 1]};
    v2f b = {bandw(kb * 4 + kb0, M), bandw(kb * 4 + kb0 + 2, M)};
    acc = wmma_f32(a, b, acc);
  }
  return acc;
}

// Vertical box sums: output row i (i=0..15) = sum of src rows hr0+i .. hr0+i+16
// at cols v0..v0+15.  Band is the A matrix, data rows are the B matrix.
__device__ __forceinline__ v8f box_v(const float* src, int ss, int hr0, int v0,
                                     int lane) {
  const int M   = lane & 15;            // A band row (output row index)
  const int N   = lane & 15;            // B column
  const int ko  = (lane < 16) ? 0 : 2;
  const int kb0 = (lane < 16) ? 0 : 1;
  v8f acc = {0.f, 0.f, 0.f, 0.f, 0.f, 0.f, 0.f, 0.f};
#pragma unroll
  for (int kb = 0; kb < 8; ++kb) {
    v2f a = {bandw(kb * 4 + ko, M), bandw(kb * 4 + ko + 1, M)};
    v2f b = {src[(hr0 + kb * 4 + kb0) * ss + v0 + N],
             src[(hr0 + kb * 4 + kb0 + 2) * ss + v0 + N]};
    acc = wmma_f32(a, b, acc);
  }
  return acc;
}

__global__ __launch_bounds__(512, 1) void guided_filter_fused(
    const float* __restrict__ xg, const float* __restrict__ yg,
    float* __restrict__ og, int Himg, int Wimg) {
  extern __shared__ float smem[];
  float* sx = smem + OFF_SX;
  float* sy = smem + OFF_SY;
  float* hq0 = smem + OFF_H0;
  float* hq1 = hq0 + HT * HS;
  float* hq2 = hq1 + HT * HS;
  float* hq3 = hq2 + HT * HS;
  float* sa = smem + OFF_SA;
  float* sb = smem + OFF_SB;
  float* ha = smem + OFF_HA;
  float* hb = smem + OFF_HB;

  const int tid  = (int)threadIdx.x;
  const int lane = tid & 31;
  const int wave = tid >> 5;
  const int nw   = (int)blockDim.x >> 5;

  const int tc = (int)blockIdx.x, tr = (int)blockIdx.y;
  const int C0 = tc * TILE - 2 * R;   // global col of x-tile origin
  const int R0 = tr * TILE - 2 * R;   // global row of x-tile origin
  const size_t planeOff = (size_t)blockIdx.z * (size_t)Himg * (size_t)Wimg;
  const float* xp = xg + planeOff;
  const float* yp = yg + planeOff;
  float* op = og + planeOff;

  // ---- Stage 0: async global->LDS fill of 96x96 x/y halo tiles.
  // In-bounds float4s: GLOBAL_LOAD_ASYNC_TO_LDS_B128 (no VGPR staging,
  // tracked by ASYNCcnt).  OOB halo: plain zero stores (DScnt, barrier-waited).
  {
    const int nvec = XT * (XT / 4);
    for (int i = tid; i < nvec; i += (int)blockDim.x) {
      const int rr = i / (XT / 4);
      const int c4 = (i - rr * (XT / 4)) * 4;
      const int gr = R0 + rr;
      const int gc = C0 + c4;                  // 4-aligned; all-in or all-out
      float* dx = sx + rr * XS + c4;
      float* dy = sy + rr * XS + c4;
      if (gr >= 0 && gr < Himg && gc >= 0 && gc < Wimg) {
        const size_t o = (size_t)gr * Wimg + gc;
        async_copy_b128((unsigned)(uintptr_t)dx, xp + o);
        async_copy_b128((unsigned)(uintptr_t)dy, yp + o);
      } else {
        dx[0] = 0.f; dx[1] = 0.f; dx[2] = 0.f; dx[3] = 0.f;
        dy[0] = 0.f; dy[1] = 0.f; dy[2] = 0.f; dy[3] = 0.f;
      }
    }
  }
  wait_async_zero();   // each wave drains its own async copies
  __syncthreads();     // then all waves agree LDS is filled

  // ---- Stage 1: horizontal box sums of x, y, x*y, x*x  (WMMA band matmuls)
  {
    const int tiles = (HT / 16) * (HC / 16);   // 6*5 = 30 per quantity
    for (int j = wave; j < 4 * tiles; j += nw) {
      const int q  = j / tiles;
      const int t  = j - q * tiles;
      const int r0 = (t / (HC / 16)) * 16;
      const int v0 = (t - (t / (HC / 16)) * (HC / 16)) * 16;
      v8f acc;
      float* dst;
      if (q == 0)      { acc = box_h(sx, XS, r0, v0, lane);           dst = hq0; }
      else if (q == 1) { acc = box_h(sy, XS, r0, v0, lane);           dst = hq1; }
      else if (q == 2) { acc = box_h_prod(sx, sy, XS, r0, v0, lane);  dst = hq2; }
      else             { acc = box_h_prod(sx, sx, XS, r0, v0, lane);  dst = hq3; }
      const int N  = lane & 15;
      const int mo = (lane < 16) ? 0 : 8;
#pragma unroll
      for (int m = 0; m < 8; ++m)
        dst[(r0 + mo + m) * HS + v0 + N] = acc[m];
    }
  }
  __syncthreads();

  // ---- Stage 2: vertical box sums -> means -> A, b on 80x80 region
  {
    const int tiles = (AT / 16) * (AT / 16);   // 25
    for (int j = wave; j < tiles; j += nw) {
      const int ar0 = (j / (AT / 16)) * 16;
      const int v0  = (j - (j / (AT / 16)) * (AT / 16)) * 16;
      v8f Sx  = box_v(hq0, HS, ar0, v0, lane);
      v8f Sy  = box_v(hq1, HS, ar0, v0, lane);
      v8f Sxy = box_v(hq2, HS, ar0, v0, lane);
      v8f Sxx = box_v(hq3, HS, ar0, v0, lane);
      const int N  = lane & 15;
      const int mo = (lane < 16) ? 0 : 8;
      const int gc = C0 + R + v0 + N;
      const float ccnt = cntf(gc, Wimg);
#pragma unroll
      for (int m = 0; m < 8; ++m) {
        const int ar = ar0 + mo + m;
        const int gr = R0 + R + ar;
        const float inv = 1.0f / (cntf(gr, Himg) * ccnt);
        const float mx  = Sx[m] * inv;
        const float my  = Sy[m] * inv;
        const float cov = Sxy[m] * inv - mx * my;
        const float var = Sxx[m] * inv - mx * mx;
        const float Av  = cov / (var + EPS);    // OOB pixels: window still reaches in-bounds x/y, so A,b not forced to 0
        const float Bv  = my - Av * mx;
        sa[ar * AS + v0 + N] = Av;
        sb[ar * AS + v0 + N] = Bv;
      }
    }
  }
  __syncthreads();

  // ---- Stage 3: horizontal box sums of A, b
  {
    const int tiles = (AT / 16) * (H2C / 16);  // 5*4 = 20 per quantity
    for (int j = wave; j < 2 * tiles; j += nw) {
      const int q  = j / tiles;
      const int t  = j - q * tiles;
      const int r0 = (t / (H2C / 16)) * 16;
      const int w0 = (t - (t / (H2C / 16)) * (H2C / 16)) * 16;
      v8f acc = box_h(q ? sb : sa, AS, r0, w0, lane);
      float* dst = q ? hb : ha;
      const int N  = lane & 15;
      const int mo = (lane < 16) ? 0 : 8;
#pragma unroll
      for (int m = 0; m < 8; ++m)
        dst[(r0 + mo + m) * H2S + w0 + N] = acc[m];
    }
  }
  __syncthreads();

  // ---- Stage 4: vertical box sums of HA, Hb + final combine + store
  {
    const int tiles = (TILE / 16) * (TILE / 16);  // 16
    for (int j = wave; j < tiles; j += nw) {
      const int or0 = (j / (TILE / 16)) * 16;
      const int w0  = (j - (j / (TILE / 16)) * (TILE / 16)) * 16;
      v8f SA = box_v(ha, H2S, or0, w0, lane);
      v8f SB = box_v(hb, H2S, or0, w0, lane);
      const int N  = lane & 15;
      const int mo = (lane < 16) ? 0 : 8;
      const int gc = tc * TILE + w0 + N;
      const float ccnt = cntf(gc, Wimg);
#pragma unroll
      for (int m = 0; m < 8; ++m) {
        const int orow = or0 + mo + m;
        const int gr = tr * TILE + orow;
        const float inv = 1.0f / (cntf(gr, Himg) * ccnt);
        const float meanA = SA[m] * inv;
        const float meanB = SB[m] * inv;
        const float xv = sx[(orow + 2 * R) * XS + (w0 + N + 2 * R)];
        op[(size_t)gr * Wimg + gc] = meanA * xv + meanB;
      }
    }
  }
}

extern "C" void kernel_launch(void* const* d_in, const int* in_sizes, int n_in,
                              void* d_out, int out_size, void* d_ws,
                              size_t ws_size, hipStream_t stream) {
  (void)d_ws; (void)ws_size; (void)n_in; (void)out_size;
  const float* x = (const float*)d_in[0];
  const float* y = (const float*)d_in[1];
  float* out = (float*)d_out;

  const int Himg = 1024, Wimg = 1024;
  const int planes = in_sizes[0] / (Himg * Wimg);  // 8*3 = 24

  const size_t smem = (size_t)SMEM_FLOATS * sizeof(float);  // ~286 KB dynamic LDS
  hipFuncSetAttribute((const void*)guided_filter_fused,
                      hipFuncAttributeMaxDynamicSharedMemorySize, (int)smem);

  dim3 grid(Wimg / TILE, Himg / TILE, planes);
  dim3 block(512);
  guided_filter_fused<<<grid, block, smem, stream>>>(x, y, out, Himg, Wimg);
}